// TextVNet_22058952032405
// MI455X (gfx1250) — compile-verified
//
#include <hip/hip_runtime.h>
#include <hip/hip_bf16.h>
#include <math.h>

// ---- CDNA5 WMMA vector types -------------------------------------------------
typedef __attribute__((ext_vector_type(2))) float v2f;
typedef __attribute__((ext_vector_type(8))) float v8f;

// Problem constants from the reference (asserted by launch-side arithmetic).
#define D2C   2048   // 2*D embed dim
#define DHALF 1024   // per-head dim
#define TPB   256

// ============================================================================
// Phase 1: dual-head batched GEMV via V_WMMA_F32_16X16X4_F32.
// Each wave owns 16 actions of one batch row. K runs over the full 2048 dims;
// B-matrix column 0 carries s[:D] (zeroed for k>=D), column 1 carries s[D:]
// (zeroed for k<D), so a single accumulation loop yields logits1 in C[:,0]
// and logits2 in C[:,1].
//
// A-matrix ISA layout (32-bit, 16x4): lane m (and m+16) hold row m;
//   lanes 0-15:  VGPR0=K0, VGPR1=K1   -> float2 at row + k + 0
//   lanes 16-31: VGPR0=K2, VGPR1=K3   -> float2 at row + k + 2
// B-matrix (4x16): lane n holds column n; VGPR v / half h holds K = 2h + v,
// mirroring the A striping -> float2 of s at (k + 2h) per lane, masked by n.
// ============================================================================
__global__ __launch_bounds__(TPB) void wmma_logits_kernel(
    const float* __restrict__ s_embed,     // [B, 2048]
    const float* __restrict__ a_emb,       // [B, A, 2048]
    float* __restrict__ logitsWS,          // [2, B, A]
    int B, int A)
{
    __shared__ float sL[D2C];

    const int tilesPerRow = A / 128;               // 128 actions per block
    const int b      = blockIdx.x / tilesPerRow;
    const int wgBase = (blockIdx.x % tilesPerRow) * 128;

    const int tid  = threadIdx.x;
    const int wave = tid >> 5;
    const int lane = tid & 31;
    const int m    = lane & 15;    // row (A) / column (B,C) index
    const int h    = lane >> 4;    // lane-half

    // Stage the state embedding (8 KB) in LDS once per workgroup.
    for (int i = tid; i < D2C; i += TPB)
        sL[i] = s_embed[(size_t)b * D2C + i];
    __syncthreads();

    const int aBase = wgBase + wave * 16;
    const float* rowPtr =
        a_emb + ((size_t)b * A + (size_t)(aBase + m)) * D2C + 2 * h;
    const float* sPtr = &sL[2 * h];

    const bool isCol0 = (m == 0);
    const bool isCol1 = (m == 1);

    v8f acc = {};
    #pragma unroll 4
    for (int k = 0; k < D2C; k += 4) {
        v2f a  = *(const v2f*)(rowPtr + k);        // A: 16 actions x K4 chunk
        v2f sv = *(const v2f*)(sPtr + k);          // s values for this lane's K
        const bool sel = (k < DHALF) ? isCol0 : isCol1;   // k<DHALF is wave-uniform
        v2f bm;
        bm.x = sel ? sv.x : 0.0f;
        bm.y = sel ? sv.y : 0.0f;
        acc = __builtin_amdgcn_wmma_f32_16x16x4_f32(
            /*neg_a=*/false, a, /*neg_b=*/false, bm,
            /*c_mod=*/(short)0, acc, /*reuse_a=*/false, /*reuse_b=*/false);
    }

    // C layout: VGPR r -> M = r (lanes 0-15) / r+8 (lanes 16-31); N = lane%16.
    // Column 0 (lanes 0,16) = head-1 logits, column 1 (lanes 1,17) = head-2.
    if (m < 2) {
        const int head = m;
        float* dst = logitsWS + ((size_t)head * B + b) * A + aBase + h * 8;
        #pragma unroll
        for (int r = 0; r < 8; ++r)
            dst[r] = acc[r];
    }
}

// ============================================================================
// Phase 2: per (head, batch) row of 2048 logits:
//   * top-5 by raw logit (5 rounds of argmax with exclusion; lowest-index
//     tie-break matches lax.top_k's stable ordering)
//   * s = max over ALL of logits/alpha (reference stabilizes unmasked)
//   * sum exp only over picked-5 AND available; clamp 1e-10
//   * out = alpha * (s + log(sum))
// ============================================================================
__global__ __launch_bounds__(TPB) void topk_lse_kernel(
    const float* __restrict__ logitsWS,    // [2, B, A]
    const int*   __restrict__ avail,       // [B, A]
    const float* __restrict__ alpha_p,     // [1]
    float* __restrict__ out,               // [2*B]  (v1 then v2)
    int B, int A)
{
    __shared__ float xs[D2C];      // A <= 2048
    __shared__ float redv[TPB];
    __shared__ int   redi[TPB];
    __shared__ int   picked[8];

    const int row  = blockIdx.x;           // head*B + b
    const int head = row / B;
    const int b    = row - head * B;
    const int tid  = threadIdx.x;

    const float alpha     = alpha_p[0];
    const float inv_alpha = 1.0f / alpha;

    const float* lp = logitsWS + (size_t)row * A;
    for (int i = tid; i < A; i += TPB) xs[i] = lp[i];
    __syncthreads();

    // ---- top-5 via 5 argmax rounds (on raw logits) ----
    for (int it = 0; it < 5; ++it) {
        float bv = -INFINITY;
        int   bi = A;
        for (int i = tid; i < A; i += TPB) {
            bool skip = false;
            for (int j = 0; j < it; ++j) skip |= (picked[j] == i);
            if (skip) continue;
            float v = xs[i];
            if (v > bv || (v == bv && i < bi)) { bv = v; bi = i; }
        }
        redv[tid] = bv; redi[tid] = bi;
        __syncthreads();
        for (int s = TPB / 2; s > 0; s >>= 1) {
            if (tid < s) {
                float ov = redv[tid + s]; int oi = redi[tid + s];
                if (ov > redv[tid] || (ov == redv[tid] && oi < redi[tid])) {
                    redv[tid] = ov; redi[tid] = oi;
                }
            }
            __syncthreads();
        }
        if (tid == 0) picked[it] = redi[0];
        __syncthreads();
    }

    // ---- unmasked max of x = logits/alpha ----
    float lm = -INFINITY;
    for (int i = tid; i < A; i += TPB) lm = fmaxf(lm, xs[i] * inv_alpha);
    redv[tid] = lm;
    __syncthreads();
    for (int s = TPB / 2; s > 0; s >>= 1) {
        if (tid < s) redv[tid] = fmaxf(redv[tid], redv[tid + s]);
        __syncthreads();
    }
    const float smax = redv[0];
    __syncthreads();

    // ---- masked exp-sum: only the <=5 picked indices can be in the mask ----
    float contrib = 0.0f;
    if (tid < 5) {
        int i = picked[tid];
        if (avail[(size_t)b * A + i] != 0)
            contrib = expf(xs[i] * inv_alpha - smax);
    }
    redv[tid] = contrib;
    __syncthreads();
    if (tid == 0) {
        float ssum = 0.0f;
        #pragma unroll
        for (int j = 0; j < 5; ++j) ssum += redv[j];
        ssum = fmaxf(ssum, 1e-10f);
        out[row] = alpha * (smax + logf(ssum));
    }
}

// ============================================================================
extern "C" void kernel_launch(void* const* d_in, const int* in_sizes, int n_in,
                              void* d_out, int out_size, void* d_ws, size_t ws_size,
                              hipStream_t stream) {
    const float* s_embed = (const float*)d_in[0];
    const float* a_emb   = (const float*)d_in[1];
    const int*   avail   = (const int*)d_in[2];
    const float* alpha   = (const float*)d_in[3];
    float* out = (float*)d_out;

    // Recover dims: in_sizes = { B*D2, B*A*D2, B*A, 1 }
    const long long BD2  = in_sizes[0];
    const long long BAD2 = in_sizes[1];
    const long long BA   = in_sizes[2];
    const int A  = (int)(BAD2 / BD2);   // 2048
    const int Bn = (int)(BA / A);       // 32

    float* logitsWS = (float*)d_ws;     // 2*B*A floats = 512 KB

    dim3 grid1(Bn * (A / 128));
    wmma_logits_kernel<<<grid1, TPB, 0, stream>>>(s_embed, a_emb, logitsWS, Bn, A);

    dim3 grid2(2 * Bn);
    topk_lse_kernel<<<grid2, TPB, 0, stream>>>(logitsWS, avail, alpha, out, Bn, A);
}